// Selective_SA_Layer_73177652789696
// MI455X (gfx1250) — compile-verified
//
#include <hip/hip_runtime.h>
#include <math.h>

#define BB 16
#define CC 256
#define NN 2048
#define DD 64
#define NTOP 1024
#define BN_EPS 1e-5f
#define ATT_LD 2064   // 2048 + 16 pad to break LDS bank alignment

typedef float v2f __attribute__((ext_vector_type(2)));
typedef float v8f __attribute__((ext_vector_type(8)));

// V_WMMA_F32_16X16X4_F32:  D(16x16,f32) = A(16x4,f32) * B(4x16,f32) + C
// A frag: lane<16 -> row M=lane, K={0,1}; lane>=16 -> row M=lane-16, K={2,3}
// B frag: lane<16 -> col N=lane, K={0,1}; lane>=16 -> col N=lane-16, K={2,3}
// C/D:    vgpr r -> M = r + (lane>=16 ? 8 : 0), N = lane&15
__device__ __forceinline__ v8f wmma_f32(v2f a, v2f b, v8f c) {
  return __builtin_amdgcn_wmma_f32_16x16x4_f32(false, a, false, b, (short)0, c,
                                               false, false);
}

// ---------------- K0: w_eff[c] = sum_k W_score[k] * W_map[k,c] ----------------
__global__ void k_weff(const float* __restrict__ W_map,
                       const float* __restrict__ W_score,
                       float* __restrict__ w_eff) {
  int c = threadIdx.x;
  float s = 0.f;
  for (int k = 0; k < CC; ++k) s += W_score[k] * W_map[k * CC + c];
  w_eff[c] = s;
}

// ---------------- K1: q_map[b,n] = w_eff . x[b,:,n]; zero colsum --------------
__global__ void k_qmap(const float* __restrict__ x,
                       const float* __restrict__ w_eff,
                       float* __restrict__ q_map, float* __restrict__ colsum) {
  int tid = blockIdx.x * blockDim.x + threadIdx.x;  // 0..32767
  int b = tid >> 11, n = tid & (NN - 1);
  const float* xb = x + (size_t)b * CC * NN + n;
  float s = 0.f;
  for (int c = 0; c < CC; ++c) s += w_eff[c] * xb[(size_t)c * NN];
  q_map[tid] = s;
  colsum[tid] = 0.f;
}

// ---------------- K2: per-batch top-1024 via bitonic sort in LDS --------------
__global__ void __launch_bounds__(1024) k_topk(const float* __restrict__ q_map,
                                               int* __restrict__ idx_out) {
  __shared__ float sv[NN];
  __shared__ int si[NN];
  int b = blockIdx.x;
  int t = threadIdx.x;
  sv[t] = q_map[b * NN + t];               si[t] = t;
  sv[t + 1024] = q_map[b * NN + t + 1024]; si[t + 1024] = t + 1024;
  __syncthreads();
  for (unsigned k = 2; k <= NN; k <<= 1) {
    for (unsigned j = k >> 1; j > 0; j >>= 1) {
      unsigned i = (((unsigned)t & ~(j - 1)) << 1) | ((unsigned)t & (j - 1));
      unsigned p = i | j;
      bool desc = ((i & k) == 0);
      float vi = sv[i], vp = sv[p];
      if ((vi < vp) == desc) {
        sv[i] = vp; sv[p] = vi;
        int ti = si[i]; si[i] = si[p]; si[p] = ti;
      }
      __syncthreads();
    }
  }
  idx_out[b * NTOP + t] = si[t];
}

// ---------------- WMMA GEMM: out[b,o,n] = W[o,:] . X[b,:,n] (+bias) -----------
// one wave -> 16(o) x 64(n) tile, A-frag reused across 4 n-subtiles
__global__ void __launch_bounds__(256)
k_gemm_wx(const float* __restrict__ W, const float* __restrict__ X,
          const float* __restrict__ bias, float* __restrict__ out,
          int O, int Cd, int N) {
  int lane = threadIdx.x & 31;
  int wave = threadIdx.x >> 5;
  int tile = blockIdx.x * 8 + wave;
  int nT = N >> 6;
  int oT = O >> 4;
  int per_b = oT * nT;
  int bi = tile / per_b;
  int rem = tile % per_b;
  int ot = rem / nT;
  int nt = rem % nT;
  const float* Xb = X + (size_t)bi * Cd * N;
  int mrow = ot * 16 + (lane & 15);
  int koff = (lane >> 4) << 1;
  int ncol = nt * 64 + (lane & 15);
  v8f acc0 = {}, acc1 = {}, acc2 = {}, acc3 = {};
  for (int k = 0; k < Cd; k += 4) {
    v2f av = *(const v2f*)(W + (size_t)mrow * Cd + k + koff);
    const float* xr0 = Xb + (size_t)(k + koff) * N + ncol;
    const float* xr1 = xr0 + N;
    v2f b0, b1, b2, b3;
    b0.x = xr0[0];  b0.y = xr1[0];
    b1.x = xr0[16]; b1.y = xr1[16];
    b2.x = xr0[32]; b2.y = xr1[32];
    b3.x = xr0[48]; b3.y = xr1[48];
    acc0 = wmma_f32(av, b0, acc0);
    acc1 = wmma_f32(av, b1, acc1);
    acc2 = wmma_f32(av, b2, acc2);
    acc3 = wmma_f32(av, b3, acc3);
  }
  int rbase = ot * 16 + ((lane >> 4) << 3);
  float* ob = out + (size_t)bi * O * N;
  for (int r = 0; r < 8; ++r) {
    int row = rbase + r;
    float bv = bias ? bias[row] : 0.f;
    float* orow = ob + (size_t)row * N + nt * 64 + (lane & 15);
    orow[0]  = acc0[r] + bv;
    orow[16] = acc1[r] + bv;
    orow[32] = acc2[r] + bv;
    orow[48] = acc3[r] + bv;
  }
}

// ---------------- x_q[b,d,m] = W_qk[d,:] . x[b,:,idx[b,m]] (gathered B) -------
__global__ void __launch_bounds__(256)
k_gemm_xq(const float* __restrict__ Wqk, const float* __restrict__ x,
          const int* __restrict__ idx, float* __restrict__ xq) {
  int lane = threadIdx.x & 31;
  int wave = threadIdx.x >> 5;
  int tile = blockIdx.x * 8 + wave;  // 1024 = 16b * 4ot * 16nt
  int bi = tile >> 6;
  int rem = tile & 63;
  int ot = rem >> 4;
  int nt = rem & 15;
  int mrow = ot * 16 + (lane & 15);
  int koff = (lane >> 4) << 1;
  const int* ib = idx + bi * NTOP + nt * 64 + (lane & 15);
  int g0 = ib[0], g1 = ib[16], g2 = ib[32], g3 = ib[48];
  v8f acc0 = {}, acc1 = {}, acc2 = {}, acc3 = {};
  const float* xb = x + (size_t)bi * CC * NN;
  for (int k = 0; k < CC; k += 4) {
    v2f av = *(const v2f*)(Wqk + (size_t)mrow * CC + k + koff);
    const float* xr0 = xb + (size_t)(k + koff) * NN;
    const float* xr1 = xr0 + NN;
    v2f b0, b1, b2, b3;
    b0.x = xr0[g0]; b0.y = xr1[g0];
    b1.x = xr0[g1]; b1.y = xr1[g1];
    b2.x = xr0[g2]; b2.y = xr1[g2];
    b3.x = xr0[g3]; b3.y = xr1[g3];
    acc0 = wmma_f32(av, b0, acc0);
    acc1 = wmma_f32(av, b1, acc1);
    acc2 = wmma_f32(av, b2, acc2);
    acc3 = wmma_f32(av, b3, acc3);
  }
  int rbase = ot * 16 + ((lane >> 4) << 3);
  float* ob = xq + (size_t)bi * DD * NTOP;
  for (int r = 0; r < 8; ++r) {
    float* orow = ob + (size_t)(rbase + r) * NTOP + nt * 64 + (lane & 15);
    orow[0] = acc0[r]; orow[16] = acc1[r]; orow[32] = acc2[r]; orow[48] = acc3[r];
  }
}

// ------- softmax row stats (online) + query-axis colsum via f32 atomics -------
__global__ void __launch_bounds__(256)
k_softstats(const float* __restrict__ xq, const float* __restrict__ xk,
            float* __restrict__ rowmax, float* __restrict__ rowsum,
            float* __restrict__ colsum) {
  int lane = threadIdx.x & 31;
  int wave = threadIdx.x >> 5;
  int tile = blockIdx.x * 8 + wave;  // 1024 = 16b * 64 m-tiles
  int bi = tile >> 6;
  int mt = tile & 63;
  int koff = (lane >> 4) << 1;
  int mglob = mt * 16 + (lane & 15);
  const float* xqb = xq + (size_t)bi * DD * NTOP;
  v2f af[16];
#pragma unroll
  for (int kk = 0; kk < 16; ++kk) {
    int k = kk * 4 + koff;
    af[kk].x = xqb[(size_t)k * NTOP + mglob];
    af[kk].y = xqb[(size_t)(k + 1) * NTOP + mglob];
  }
  const float* xkb = xk + (size_t)bi * DD * NN;
  float mrun[8], srun[8];
#pragma unroll
  for (int r = 0; r < 8; ++r) { mrun[r] = -INFINITY; srun[r] = 0.f; }
  for (int nt = 0; nt < NN / 16; ++nt) {
    int ncol = nt * 16 + (lane & 15);
    v8f e = {};
#pragma unroll
    for (int kk = 0; kk < 16; ++kk) {
      int k = kk * 4 + koff;
      v2f bf;
      bf.x = xkb[(size_t)k * NN + ncol];
      bf.y = xkb[(size_t)(k + 1) * NN + ncol];
      e = wmma_f32(af[kk], bf, e);
    }
#pragma unroll
    for (int r = 0; r < 8; ++r) {
      float v = e[r];
      float mn = fmaxf(mrun[r], v);
      srun[r] = srun[r] * __expf(mrun[r] - mn) + __expf(v - mn);
      mrun[r] = mn;
    }
  }
  // merge (max,sum) across the 16 lanes that share each row group
#pragma unroll
  for (int mask = 1; mask <= 8; mask <<= 1) {
#pragma unroll
    for (int r = 0; r < 8; ++r) {
      float mo = __shfl_xor(mrun[r], mask, 32);
      float so = __shfl_xor(srun[r], mask, 32);
      float mn = fmaxf(mrun[r], mo);
      srun[r] = srun[r] * __expf(mrun[r] - mn) + so * __expf(mo - mn);
      mrun[r] = mn;
    }
  }
  if ((lane & 15) == 0) {
    int rb = bi * NTOP + mt * 16 + ((lane >> 4) << 3);
    for (int r = 0; r < 8; ++r) { rowmax[rb + r] = mrun[r]; rowsum[rb + r] = srun[r]; }
  }
  // second sweep: accumulate colsum[b,n] = sum_m softmax(e)[m,n]
  for (int nt = 0; nt < NN / 16; ++nt) {
    int ncol = nt * 16 + (lane & 15);
    v8f e = {};
#pragma unroll
    for (int kk = 0; kk < 16; ++kk) {
      int k = kk * 4 + koff;
      v2f bf;
      bf.x = xkb[(size_t)k * NN + ncol];
      bf.y = xkb[(size_t)(k + 1) * NN + ncol];
      e = wmma_f32(af[kk], bf, e);
    }
    float cp = 0.f;
#pragma unroll
    for (int r = 0; r < 8; ++r) cp += __expf(e[r] - mrun[r]) / srun[r];
    cp += __shfl_xor(cp, 16, 32);
    if (lane < 16) atomicAdd(&colsum[bi * NN + ncol], cp);
  }
}

// ------- pass C: attn strip in LDS, then x_r = x_v @ attn^T; emit dsel --------
__global__ void __launch_bounds__(256)
k_attn_apply(const float* __restrict__ xq, const float* __restrict__ xk,
             const float* __restrict__ xv, const float* __restrict__ x,
             const float* __restrict__ rowmax, const float* __restrict__ rowsum,
             const float* __restrict__ colsum, const int* __restrict__ idx,
             float* __restrict__ dsel) {
  extern __shared__ float attn_s[];  // 16 * ATT_LD floats (132 KB)
  int lane = threadIdx.x & 31;
  int wave = threadIdx.x >> 5;
  int bi = blockIdx.x >> 6;
  int mt = blockIdx.x & 63;
  int koff = (lane >> 4) << 1;
  int lrow = (lane >> 4) << 3;  // 0 or 8
  float rmax[8], rsum[8];
#pragma unroll
  for (int r = 0; r < 8; ++r) {
    rmax[r] = rowmax[bi * NTOP + mt * 16 + lrow + r];
    rsum[r] = rowsum[bi * NTOP + mt * 16 + lrow + r];
  }
  int mglob = mt * 16 + (lane & 15);
  const float* xqb = xq + (size_t)bi * DD * NTOP;
  v2f af[16];
#pragma unroll
  for (int kk = 0; kk < 16; ++kk) {
    int k = kk * 4 + koff;
    af[kk].x = xqb[(size_t)k * NTOP + mglob];
    af[kk].y = xqb[(size_t)(k + 1) * NTOP + mglob];
  }
  const float* xkb = xk + (size_t)bi * DD * NN;
  // phase 1: this wave fills 256 columns of the 16 x 2048 attn strip
  for (int nt = wave * 16; nt < wave * 16 + 16; ++nt) {
    int ncol = nt * 16 + (lane & 15);
    v8f e = {};
#pragma unroll
    for (int kk = 0; kk < 16; ++kk) {
      int k = kk * 4 + koff;
      v2f bf;
      bf.x = xkb[(size_t)k * NN + ncol];
      bf.y = xkb[(size_t)(k + 1) * NN + ncol];
      e = wmma_f32(af[kk], bf, e);
    }
    float cs = 1e-9f + colsum[bi * NN + ncol];
#pragma unroll
    for (int r = 0; r < 8; ++r) {
      float p = __expf(e[r] - rmax[r]) / rsum[r] / cs;
      attn_s[(lrow + r) * ATT_LD + ncol] = p;
    }
  }
  __syncthreads();
  // phase 2: x_r c-tiles {wave, wave+8}; dsel = x[:,idx] - x_r
  const float* xvb = xv + (size_t)bi * CC * NN;
  const float* xb = x + (size_t)bi * CC * NN;
  int mcol = lane & 15;
  int nsel = idx[bi * NTOP + mt * 16 + mcol];
  float* db = dsel + (size_t)bi * CC * NTOP;
  for (int cti = 0; cti < 2; ++cti) {
    int ct = wave + cti * 8;
    int crow = ct * 16 + (lane & 15);
    v8f acc = {};
    for (int n = 0; n < NN; n += 4) {
      v2f av = *(const v2f*)(xvb + (size_t)crow * NN + n + koff);
      v2f bf = *(const v2f*)(&attn_s[mcol * ATT_LD + n + koff]);
      acc = wmma_f32(av, bf, acc);
    }
#pragma unroll
    for (int r = 0; r < 8; ++r) {
      int c = ct * 16 + lrow + r;
      db[(size_t)c * NTOP + mt * 16 + mcol] =
          xb[(size_t)c * NN + nsel] - acc[r];
    }
  }
}

// ------------- epilogues: residual + BN + ReLU ---------------
__global__ void k_resid_base(const float* __restrict__ x,
                             const float* __restrict__ b_trans,
                             const float* __restrict__ g,
                             const float* __restrict__ be,
                             const float* __restrict__ mu,
                             const float* __restrict__ var,
                             float* __restrict__ out) {
  int tid = blockIdx.x * blockDim.x + threadIdx.x;
  int c = (tid >> 11) & 255;
  float bn = g[c] * (b_trans[c] - mu[c]) * rsqrtf(var[c] + BN_EPS) + be[c];
  out[tid] = x[tid] + fmaxf(bn, 0.f);
}

__global__ void k_resid_sel(const float* __restrict__ x,
                            const float* __restrict__ tsel,
                            const int* __restrict__ idx,
                            const float* __restrict__ g,
                            const float* __restrict__ be,
                            const float* __restrict__ mu,
                            const float* __restrict__ var,
                            float* __restrict__ out) {
  int tid = blockIdx.x * blockDim.x + threadIdx.x;  // b*256*1024 total
  int m = tid & 1023;
  int c = (tid >> 10) & 255;
  int bi = tid >> 18;
  int n = idx[bi * NTOP + m];
  float t = tsel[tid];  // t_sel layout [b,c,m] == tid, already includes b_trans
  float bn = g[c] * (t - mu[c]) * rsqrtf(var[c] + BN_EPS) + be[c];
  size_t o = ((size_t)bi * CC + c) * NN + n;
  out[o] = x[o] + fmaxf(bn, 0.f);
}

extern "C" void kernel_launch(void* const* d_in, const int* in_sizes, int n_in,
                              void* d_out, int out_size, void* d_ws,
                              size_t ws_size, hipStream_t stream) {
  const float* x       = (const float*)d_in[0];
  const float* W_map   = (const float*)d_in[1];
  const float* W_score = (const float*)d_in[2];
  const float* W_qk    = (const float*)d_in[3];
  const float* W_v     = (const float*)d_in[4];
  const float* b_v     = (const float*)d_in[5];
  const float* W_trans = (const float*)d_in[6];
  const float* b_trans = (const float*)d_in[7];
  const float* gamma   = (const float*)d_in[8];
  const float* beta    = (const float*)d_in[9];
  const float* mu      = (const float*)d_in[10];
  const float* var     = (const float*)d_in[11];
  float* out = (float*)d_out;
  float* ws = (float*)d_ws;

  float* w_eff  = ws;                  // 256
  float* q_map  = ws + 256;            // 32768
  float* colsum = ws + 33024;          // 32768
  float* rowmax = ws + 65792;          // 16384
  float* rowsum = ws + 82176;          // 16384
  int*   idx    = (int*)(ws + 98560);  // 16384
  float* x_k    = ws + 114944;         // 16*64*2048
  float* x_q    = ws + 2212096;        // 16*64*1024
  float* x_v    = ws + 3260672;        // 16*256*2048
  float* dsel   = ws + 11649280;       // 16*256*1024
  float* t_sel  = ws + 15843584;       // 16*256*1024

  k_weff<<<1, 256, 0, stream>>>(W_map, W_score, w_eff);
  k_qmap<<<128, 256, 0, stream>>>(x, w_eff, q_map, colsum);
  k_topk<<<16, 1024, 0, stream>>>(q_map, idx);
  k_gemm_wx<<<256, 256, 0, stream>>>(W_qk, x, nullptr, x_k, DD, CC, NN);
  k_gemm_wx<<<1024, 256, 0, stream>>>(W_v, x, b_v, x_v, CC, CC, NN);
  k_gemm_xq<<<128, 256, 0, stream>>>(W_qk, x, idx, x_q);
  k_softstats<<<128, 256, 0, stream>>>(x_q, x_k, rowmax, rowsum, colsum);
  k_attn_apply<<<1024, 256, 16 * ATT_LD * sizeof(float), stream>>>(
      x_q, x_k, x_v, x, rowmax, rowsum, colsum, idx, dsel);
  k_gemm_wx<<<512, 256, 0, stream>>>(W_trans, dsel, b_trans, t_sel, CC, CC, NTOP);
  k_resid_base<<<32768, 256, 0, stream>>>(x, b_trans, gamma, beta, mu, var, out);
  k_resid_sel<<<16384, 256, 0, stream>>>(x, t_sel, idx, gamma, beta, mu, var, out);
}